// IIRFilter_63307817943156
// MI455X (gfx1250) — compile-verified
//
#include <hip/hip_runtime.h>
#include <hip/hip_bf16.h>

// ---------------------------------------------------------------------------
// filtfilt (order-4 Butterworth bandpass, 9 taps, 8 states) on (512, 65536) f32
// Chunked parallel linear scan:
//   Phase A: per (16-row group, chunk) wave, zero-state filtering via WMMA
//            block recurrence  [Y; z'] = Mpad(32x32) @ [X; z; 0]  (K=24 live)
//   Phase B: per-row scan of chunk boundary states  z_{c+1} = A^L z_c + d_c
//   Phase C: superposition correction  y[t] += W[t%L] . z_in[chunk]
// Backward pass = same kernels specialized with reversed indexing, in place.
// ---------------------------------------------------------------------------

typedef float v2f __attribute__((ext_vector_type(2)));
typedef float v4f __attribute__((ext_vector_type(4)));
typedef float v8f __attribute__((ext_vector_type(8)));

#define NROWS   512
#define STATES  8
#define CHUNK   1024          // L: samples per chunk -> 2048 phase-A waves
#define BLK     16            // samples per WMMA block step

// ---------------------------------------------------------------------------
// Table builder: one wave; lane j simulates basis channel j.
//   Mtab[32][32]  block map (rows 0..15 = y, 16..23 = z'; cols 0..15 = x,
//                 16..23 = z)
//   Wtab[L][8]    y response at offset t to unit initial state e_j
//   ALtab[8][8]   state after L zero-input steps from unit state e_j
// ---------------------------------------------------------------------------
__global__ void __launch_bounds__(32) build_tables(const float* __restrict__ bc,
                                                   const float* __restrict__ ac,
                                                   float* ws, int L) {
  const int l = threadIdx.x;
  const float a0 = ac[0];
  const float b0 = bc[0] / a0;
  float bt[STATES], at[STATES];
  for (int i = 0; i < STATES; ++i) { bt[i] = bc[i + 1] / a0; at[i] = ac[i + 1] / a0; }

  float* Mtab  = ws;                 // 32*32
  float* Wtab  = ws + 1024;          // L*8
  float* ALtab = Wtab + (long)L * 8; // 8*8

  for (int i = l; i < 1024; i += 32) Mtab[i] = 0.f;
  __syncthreads();

  if (l < 24) {
    float z[STATES];
    for (int i = 0; i < STATES; ++i) z[i] = (l >= 16 && (l - 16) == i) ? 1.f : 0.f;
    for (int t = 0; t < BLK; ++t) {
      float xv = (l < 16 && t == l) ? 1.f : 0.f;
      float yv = b0 * xv + z[0];
      float nz[STATES];
      for (int i = 0; i < STATES; ++i)
        nz[i] = bt[i] * xv - at[i] * yv + (i < STATES - 1 ? z[i + 1] : 0.f);
      for (int i = 0; i < STATES; ++i) z[i] = nz[i];
      Mtab[t * 32 + l] = yv;
    }
    for (int i = 0; i < STATES; ++i) Mtab[(BLK + i) * 32 + l] = z[i];
  }

  if (l < STATES) {
    float z[STATES];
    for (int i = 0; i < STATES; ++i) z[i] = (i == l) ? 1.f : 0.f;
    for (int t = 0; t < L; ++t) {
      float yv = z[0];
      Wtab[(long)t * 8 + l] = yv;
      float nz[STATES];
      for (int i = 0; i < STATES; ++i)
        nz[i] = -at[i] * yv + (i < STATES - 1 ? z[i + 1] : 0.f);
      for (int i = 0; i < STATES; ++i) z[i] = nz[i];
    }
    for (int i = 0; i < STATES; ++i) ALtab[i * 8 + l] = z[i];
  }
}

// ---------------------------------------------------------------------------
// Phase A: each wave handles 16 rows x one chunk. DIR is a compile-time
// direction; all global access goes through per-lane pointers advanced by a
// constant stride so loads/stores use immediate instruction offsets.
// Per 16-sample block: 12 x v_wmma_f32_16x16x4_f32; state carried through LDS.
// Wave-uniform control flow: EXEC is all-ones at every WMMA.
// ---------------------------------------------------------------------------
template <int DIR>
__global__ void __launch_bounds__(256) phaseA(const float* src, float* dst,
                                              const float* __restrict__ Mtab,
                                              float* dstate,
                                              int T, int L, int NC) {
  __shared__ __align__(16) float Mlds[32 * 32];
  __shared__ __align__(16) float zbuf[8][32][8];

  const int tid = threadIdx.x;
  for (int i = tid; i < 1024; i += 256) Mlds[i] = Mtab[i];
  __syncthreads();

  const int wave = tid >> 5;
  const int lane = tid & 31;
  const int n    = lane & 15;          // data row within the 16-row group
  const int hi   = (lane >> 4) & 1;    // lane half
  const int kb   = hi * 2;             // K sub-offset per assumed A/B layout

  const int task = blockIdx.x * 8 + wave;   // tasks = (512/16) * NC, exact fit
  const int rg   = task / NC;
  const int c    = task % NC;
  const int row  = rg * 16 + n;
  const int tchunk = c * L;

  // Direction-mirrored row base pointers: logical time t lives at p +/- t.
  const float* ps = src + (long)row * T + (DIR ? (T - 1) : 0);
  float*       pd = dst + (long)row * T + (DIR ? (T - 1) : 0);

  // Per-lane streaming pointers (advance by +/-16 floats per block step;
  // granule kk sits at a constant +/-4*kk float offset -> immediate ioffset).
  const float* pX;
  float*       pY;
  if (DIR == 0) {
    pX = ps + (tchunk + kb);
    pY = pd + (tchunk + hi * 8);
  } else {
    pX = ps - (tchunk + kb + 1);
    pY = pd - (tchunk + hi * 8 + 7);
  }

  // Preload A-operand tiles (constant across the whole chunk):
  // tile (tr, kk): lane<16 holds M=lane, K={4kk, 4kk+1}; lane>=16 M=lane-16,
  // K={4kk+2, 4kk+3}.
  v2f aM[2][6];
  for (int tr = 0; tr < 2; ++tr)
    for (int kk = 0; kk < 6; ++kk)
      aM[tr][kk] = *(const v2f*)&Mlds[(tr * 16 + n) * 32 + kk * 4 + kb];

  // zero-state init of carried state
  for (int j = 0; j < STATES; ++j) zbuf[wave][lane][j] = 0.f;
  asm volatile("s_wait_dscnt 0x0" ::: "memory");
  __builtin_amdgcn_wave_barrier();

  const int NB = L >> 4;

  for (int blk = 0; blk < NB; ++blk) {
    // B-operand tiles for K=0..15: one aligned 8-byte load each
    // (reverse direction loads the mirrored pair and swaps in registers).
    v2f bx[4];
#pragma unroll
    for (int kk = 0; kk < 4; ++kk) {
      if (DIR == 0) {
        bx[kk] = *(const v2f*)(pX + 4 * kk);
      } else {
        v2f t = *(const v2f*)(pX - 4 * kk);
        bx[kk][0] = t[1];
        bx[kk][1] = t[0];
      }
    }

    // Unconditional speculative stream-ahead hint (DEV-scope prefetch;
    // OOB translation failures are silently dropped per ISA).
    if (DIR == 0) __builtin_prefetch(pX + BLK, 0, 1);
    else          __builtin_prefetch(pX - BLK, 0, 1);

    // B-operand tiles for K=16..23 (carried state) from LDS
    v2f bz[2];
#pragma unroll
    for (int zt = 0; zt < 2; ++zt)
      bz[zt] = *(const v2f*)&zbuf[wave][n][zt * 4 + kb];

    v8f y  = {};
    v8f zn = {};
#pragma unroll
    for (int kk = 0; kk < 4; ++kk) {
      y  = __builtin_amdgcn_wmma_f32_16x16x4_f32(false, aM[0][kk], false, bx[kk],
                                                 (short)0, y,  false, false);
      zn = __builtin_amdgcn_wmma_f32_16x16x4_f32(false, aM[1][kk], false, bx[kk],
                                                 (short)0, zn, false, false);
    }
#pragma unroll
    for (int zt = 0; zt < 2; ++zt) {
      y  = __builtin_amdgcn_wmma_f32_16x16x4_f32(false, aM[0][4 + zt], false, bz[zt],
                                                 (short)0, y,  false, false);
      zn = __builtin_amdgcn_wmma_f32_16x16x4_f32(false, aM[1][4 + zt], false, bz[zt],
                                                 (short)0, zn, false, false);
    }

    // Store zero-state outputs: lane holds 8 consecutive logical times ->
    // two aligned 16B stores (element-reversed for DIR=1).
    if (DIR == 0) {
      *(v4f*)(pY)     = v4f{y[0], y[1], y[2], y[3]};
      *(v4f*)(pY + 4) = v4f{y[4], y[5], y[6], y[7]};
    } else {
      *(v4f*)(pY)     = v4f{y[7], y[6], y[5], y[4]};
      *(v4f*)(pY + 4) = v4f{y[3], y[2], y[1], y[0]};
    }

    // New state back to LDS (lanes>=16 hold the zero padding rows — harmless)
#pragma unroll
    for (int j = 0; j < STATES; ++j) zbuf[wave][lane][j] = zn[j];
    asm volatile("s_wait_dscnt 0x0" ::: "memory");
    __builtin_amdgcn_wave_barrier();

    pX += (DIR ? -BLK : BLK);
    pY += (DIR ? -BLK : BLK);
  }

  // Chunk boundary state d = zero-state final state (for the Phase-B scan)
  if (lane < 16) {
    float* dp = dstate + ((long)row * NC + c) * 8;
    *(v4f*)(dp)     = *(const v4f*)&zbuf[wave][lane][0];
    *(v4f*)(dp + 4) = *(const v4f*)&zbuf[wave][lane][4];
  }
}

// ---------------------------------------------------------------------------
// Phase B: per-row sequential scan over chunks (8x8 matvec per chunk).
// ---------------------------------------------------------------------------
__global__ void __launch_bounds__(256) phaseB(const float* __restrict__ ALtab,
                                              const float* __restrict__ dstate,
                                              float* zin, int NC) {
  const int r = blockIdx.x * blockDim.x + threadIdx.x;
  if (r >= NROWS) return;
  float A[64];
  for (int i = 0; i < 64; ++i) A[i] = ALtab[i];
  float z[STATES] = {0.f, 0.f, 0.f, 0.f, 0.f, 0.f, 0.f, 0.f};
  const float* dp = dstate + (long)r * NC * 8;
  float* zp = zin + (long)r * NC * 8;
  for (int c = 0; c < NC; ++c) {
    for (int j = 0; j < STATES; ++j) zp[c * 8 + j] = z[j];
    float nz[STATES];
    for (int i = 0; i < STATES; ++i) {
      float acc = dp[c * 8 + i];
      for (int j = 0; j < STATES; ++j) acc += A[i * 8 + j] * z[j];
      nz[i] = acc;
    }
    for (int i = 0; i < STATES; ++i) z[i] = nz[i];
  }
}

// ---------------------------------------------------------------------------
// Phase C: superposition correction, 8 FMAs/sample (W and zin are L2-resident).
// ---------------------------------------------------------------------------
template <int DIR>
__global__ void __launch_bounds__(256) phaseC(float* y,
                                              const float* __restrict__ Wtab,
                                              const float* __restrict__ zin,
                                              int T, int L, int NC) {
  const long gid = (long)blockIdx.x * blockDim.x + threadIdx.x;
  const int r = (int)(gid / T);
  const int u = (int)(gid % T);              // logical (filter-direction) time
  const int p = DIR ? (T - 1 - u) : u;       // physical index
  const int c = u / L;
  const int s = u % L;
  const float* zp = zin + ((long)r * NC + c) * 8;
  const float* wr = Wtab + (long)s * 8;
  float acc = 0.f;
#pragma unroll
  for (int j = 0; j < STATES; ++j) acc += wr[j] * zp[j];
  y[(long)r * T + p] += acc;
}

// ---------------------------------------------------------------------------
extern "C" void kernel_launch(void* const* d_in, const int* in_sizes, int n_in,
                              void* d_out, int out_size, void* d_ws, size_t ws_size,
                              hipStream_t stream) {
  const float* x = (const float*)d_in[0];
  const float* b = (const float*)d_in[1];
  const float* a = (const float*)d_in[2];
  float* out = (float*)d_out;
  float* ws  = (float*)d_ws;

  const int B  = NROWS;
  const int T  = in_sizes[0] / B;      // 65536
  const int L  = CHUNK;                // 1024
  const int NC = T / L;                // 64

  // workspace layout (floats)
  float* Mtab  = ws;                       // 1024
  float* Wtab  = ws + 1024;                // L*8
  float* ALtab = Wtab + (long)L * 8;       // 64
  float* d1    = ALtab + 64;               // B*NC*8
  float* d2    = d1 + (long)B * NC * 8;
  float* z1    = d2 + (long)B * NC * 8;
  float* z2    = z1 + (long)B * NC * 8;

  build_tables<<<dim3(1), dim3(32), 0, stream>>>(b, a, ws, L);

  const int tasks = (B / 16) * NC;         // 2048 waves
  dim3 gA(tasks / 8), bA(256);
  dim3 gB((B + 255) / 256), bB(256);
  dim3 gC((unsigned)(((long)B * T) / 256)), bC(256);

  // forward pass
  phaseA<0><<<gA, bA, 0, stream>>>(x, out, Mtab, d1, T, L, NC);
  phaseB<<<gB, bB, 0, stream>>>(ALtab, d1, z1, NC);
  phaseC<0><<<gC, bC, 0, stream>>>(out, Wtab, z1, T, L, NC);
  // backward pass (in place, reversed indexing)
  phaseA<1><<<gA, bA, 0, stream>>>(out, out, Mtab, d2, T, L, NC);
  phaseB<<<gB, bB, 0, stream>>>(ALtab, d2, z2, NC);
  phaseC<1><<<gC, bC, 0, stream>>>(out, Wtab, z2, T, L, NC);
}